// Model_60773787238404
// MI455X (gfx1250) — compile-verified
//
#include <hip/hip_runtime.h>

#define HD 64

typedef __attribute__((ext_vector_type(2))) float v2f;
typedef __attribute__((ext_vector_type(8))) float v8f;

// ---------------------------------------------------------------- zero
__global__ void zero_f32(float* __restrict__ p, long n) {
  long i = (long)blockIdx.x * blockDim.x + threadIdx.x;
  if (i < n) p[i] = 0.0f;
}

// ---------------------------------------------------------------- degree
__global__ void degree_kernel(const int* __restrict__ dst, float* __restrict__ deg, int E) {
  int e = blockIdx.x * blockDim.x + threadIdx.x;
  if (e < E) atomicAdd(&deg[dst[e]], 1.0f);
}

// ------------------------------------------------------- scatter-add msg
// thread t handles edge e = t/16, 4 consecutive floats c = (t%16)*4.
// Gathers from xs[src] (optionally via node_id) and atomically adds into
// agg[dst]. Whole working set is L2-resident (192MB), atomics resolve in L2.
__global__ void scatter_kernel(const float* __restrict__ xs, const int* __restrict__ nid,
                               const int* __restrict__ src, const int* __restrict__ dst,
                               float* __restrict__ agg, int E) {
  int t = blockIdx.x * blockDim.x + threadIdx.x;
  int e = t >> 4, c = t & 15;
  if (e >= E) return;
  int s = src[e];
  if (nid) s = nid[s];
  const float4 v = *(const float4*)(xs + (long)s * HD + c * 4);
  float* a = agg + (long)dst[e] * HD + c * 4;
  atomicAdd(a + 0, v.x);
  atomicAdd(a + 1, v.y);
  atomicAdd(a + 2, v.z);
  atomicAdd(a + 3, v.w);
}

// ------------------------------------------------- fused SAGE dense layer
// Block = 256 threads = 8 waves; Wl/Wr staged in LDS once per block (32KB).
// One wave32 per 16-node row tile:
//   out[16x64] = act( (agg/deg) @ Wl^T + x @ Wr^T + b )
// via V_WMMA_F32_16X16X4_F32 with 4 independent accumulators (one per
// 16-col n-tile) interleaved to break the WMMA RAW chain and keep register
// pressure low (no spills).
// Operand layouts per ISA 7.12.2:
//  A 16x4 f32: lane m=lane&15 is row; khalf=lane>>4 picks K{0,1}/{2,3} -> v2f.
//  B 4x16:     lane col n=lane&15, same khalf -> contiguous W[n][k..k+1].
//  C 16x16:    VGPR r holds row r+8*khalf, col lane&15.
__global__ void sage_dense_wmma(const float* __restrict__ agg, const float* __restrict__ deg,
                                const float* __restrict__ xrow, const int* __restrict__ nid,
                                const float* __restrict__ Wl, const float* __restrict__ Wr,
                                const float* __restrict__ bias, float* __restrict__ out,
                                int n, int relu) {
  __shared__ float lWl[HD * HD];
  __shared__ float lWr[HD * HD];

  // Cooperative weight staging (all threads participate before any exit).
  for (int i = threadIdx.x; i < (HD * HD) / 4; i += blockDim.x) {
    ((float4*)lWl)[i] = ((const float4*)Wl)[i];
    ((float4*)lWr)[i] = ((const float4*)Wr)[i];
  }
  __syncthreads();

  int wave = (int)((blockIdx.x * blockDim.x + threadIdx.x) >> 5);
  int lane = threadIdx.x & 31;
  int row0 = wave << 4;
  if (row0 >= n) return;               // wave-uniform: EXEC all-1s for WMMA

  int mrow  = lane & 15;
  int khalf = lane >> 4;
  int row = row0 + mrow;
  if (row >= n) row = n - 1;           // N%16==0; safety only
  int xr = nid ? nid[row] : row;

  float inv = 1.0f / fmaxf(deg[row], 1.0f);

  const float* ap = agg  + (long)row * HD + khalf * 2;
  const float* xp = xrow + (long)xr  * HD + khalf * 2;

  v8f c0 = {}, c1 = {}, c2 = {}, c3 = {};

#pragma unroll
  for (int k = 0; k < 16; ++k) {
    int ko = k * 4;
    v2f aA, aX;
    aA.x = ap[ko] * inv;  aA.y = ap[ko + 1] * inv;
    aX.x = xp[ko];        aX.y = xp[ko + 1];

    const float* wl = lWl + mrow * HD + ko + khalf * 2;
    const float* wr = lWr + mrow * HD + ko + khalf * 2;
    v2f bl0 = *(const v2f*)(wl + 0 * 16 * HD);
    v2f bl1 = *(const v2f*)(wl + 1 * 16 * HD);
    v2f bl2 = *(const v2f*)(wl + 2 * 16 * HD);
    v2f bl3 = *(const v2f*)(wl + 3 * 16 * HD);
    c0 = __builtin_amdgcn_wmma_f32_16x16x4_f32(false, aA, false, bl0, (short)0, c0, false, false);
    c1 = __builtin_amdgcn_wmma_f32_16x16x4_f32(false, aA, false, bl1, (short)0, c1, false, false);
    c2 = __builtin_amdgcn_wmma_f32_16x16x4_f32(false, aA, false, bl2, (short)0, c2, false, false);
    c3 = __builtin_amdgcn_wmma_f32_16x16x4_f32(false, aA, false, bl3, (short)0, c3, false, false);

    v2f br0 = *(const v2f*)(wr + 0 * 16 * HD);
    v2f br1 = *(const v2f*)(wr + 1 * 16 * HD);
    v2f br2 = *(const v2f*)(wr + 2 * 16 * HD);
    v2f br3 = *(const v2f*)(wr + 3 * 16 * HD);
    c0 = __builtin_amdgcn_wmma_f32_16x16x4_f32(false, aX, false, br0, (short)0, c0, false, false);
    c1 = __builtin_amdgcn_wmma_f32_16x16x4_f32(false, aX, false, br1, (short)0, c1, false, false);
    c2 = __builtin_amdgcn_wmma_f32_16x16x4_f32(false, aX, false, br2, (short)0, c2, false, false);
    c3 = __builtin_amdgcn_wmma_f32_16x16x4_f32(false, aX, false, br3, (short)0, c3, false, false);
  }

  // Epilogue: bias (+relu) and store. C layout: row r+8*khalf, col mrow.
  v8f cc[4] = {c0, c1, c2, c3};
#pragma unroll
  for (int nt = 0; nt < 4; ++nt) {
    int n0 = nt << 4;
    float bv = bias[n0 + mrow];
#pragma unroll
    for (int r = 0; r < 8; ++r) {
      int orow = row0 + r + khalf * 8;
      float v = cc[nt][r] + bv;
      if (relu) v = fmaxf(v, 0.0f);
      if (orow < n) out[(long)orow * HD + n0 + mrow] = v;
    }
  }
}

// ------------------------------------------------------ per-edge dot score
// 8 lanes per edge, each lane 8 floats (2x float4), subwarp shuffle reduce.
__global__ void edge_dot_kernel(const float* __restrict__ x, const int* __restrict__ src,
                                const int* __restrict__ dst, float* __restrict__ out, int E) {
  int t = blockIdx.x * blockDim.x + threadIdx.x;
  int e = t >> 3, l = t & 7;
  if (e >= E) return;
  const float4* ps = (const float4*)(x + (long)src[e] * HD) + l * 2;
  const float4* pd = (const float4*)(x + (long)dst[e] * HD) + l * 2;
  float4 s0 = ps[0], s1 = ps[1], d0 = pd[0], d1 = pd[1];
  float acc = s0.x * d0.x + s0.y * d0.y + s0.z * d0.z + s0.w * d0.w
            + s1.x * d1.x + s1.y * d1.y + s1.z * d1.z + s1.w * d1.w;
  acc += __shfl_xor(acc, 4, 8);
  acc += __shfl_xor(acc, 2, 8);
  acc += __shfl_xor(acc, 1, 8);
  if (l == 0) out[e] = acc;
}

// ----------------------------------------------------------------- launch
extern "C" void kernel_launch(void* const* d_in, const int* in_sizes, int n_in,
                              void* d_out, int out_size, void* d_ws, size_t ws_size,
                              hipStream_t stream) {
  const float* emb = (const float*)d_in[0];
  const float* Wl1 = (const float*)d_in[1];
  const float* Wr1 = (const float*)d_in[2];
  const float* b1  = (const float*)d_in[3];
  const float* Wl2 = (const float*)d_in[4];
  const float* Wr2 = (const float*)d_in[5];
  const float* b2  = (const float*)d_in[6];
  const int*   nid = (const int*)d_in[7];
  const int*   ei  = (const int*)d_in[8];

  const int N = in_sizes[7];
  const int E = in_sizes[8] / 2;
  const int* src = ei;
  const int* dst = ei + E;

  float* deg = (float*)d_ws;                 // [N]
  float* agg = deg + N;                      // [N*64] (reused for both layers)
  float* h1  = agg + (size_t)N * HD;         // [N*64]
  float* x2  = h1  + (size_t)N * HD;         // [N*64]
  float* outF = (float*)d_out;               // [E]

  auto cdiv = [](long a, long b) { return (int)((a + b - 1) / b); };
  const int BS = 256;
  const int waves = (N + 15) / 16;

  // deg and agg are contiguous -> one zeroing pass
  zero_f32<<<cdiv((long)N * (HD + 1), BS), BS, 0, stream>>>(deg, (long)N * (HD + 1));
  degree_kernel<<<cdiv(E, BS), BS, 0, stream>>>(dst, deg, E);

  // Layer 1: scatter emb[node_id[src]] -> agg, then fused WMMA dense + relu
  scatter_kernel<<<cdiv((long)E * 16, BS), BS, 0, stream>>>(emb, nid, src, dst, agg, E);
  sage_dense_wmma<<<cdiv((long)waves * 32, BS), BS, 0, stream>>>(
      agg, deg, emb, nid, Wl1, Wr1, b1, h1, N, 1);

  // Layer 2
  zero_f32<<<cdiv((long)N * HD, BS), BS, 0, stream>>>(agg, (long)N * HD);
  scatter_kernel<<<cdiv((long)E * 16, BS), BS, 0, stream>>>(h1, nullptr, src, dst, agg, E);
  sage_dense_wmma<<<cdiv((long)waves * 32, BS), BS, 0, stream>>>(
      agg, deg, h1, nullptr, Wl2, Wr2, b2, x2, N, 0);

  // Edge scores
  edge_dot_kernel<<<cdiv((long)E * 8, BS), BS, 0, stream>>>(x2, src, dst, outF, E);
}